// PowerFlowConsistency_38010460570139
// MI455X (gfx1250) — compile-verified
//
#include <hip/hip_runtime.h>
#include <hip/hip_bf16.h>

#define EPS_F   1e-6f
#define V_LO_F  0.95f
#define V_HI_F  1.05f

typedef __attribute__((ext_vector_type(2))) float v2f;
typedef __attribute__((ext_vector_type(8))) float v8f;

// ---------------------------------------------------------------------------
// Zero the workspace accumulators (interleaved pq_flow, scalar acc).
// ---------------------------------------------------------------------------
__global__ void pf_zero_kernel(float* __restrict__ ws, int n) {
    int i = blockIdx.x * blockDim.x + threadIdx.x;
    if (i < n) ws[i] = 0.0f;
}

// ---------------------------------------------------------------------------
// Edge pass: non-temporal streaming reads (read-once data, keep L2 for the
// atomic target region) + random gather of node voltage components, then
// scatter-add p_e/q_e with hardware global_atomic_add_f32 (no-return form).
// pq_flow is interleaved so both atomics per endpoint hit one cacheline.
// Grid-stride loop with global_prefetch_b8 on next iteration's stream data.
// ---------------------------------------------------------------------------
__global__ __launch_bounds__(256) void pf_edge_kernel(
    const float2* __restrict__ nf2,        // node_features viewed as float2 pairs
    const long long* __restrict__ src_idx, // edge_index row 0
    const long long* __restrict__ dst_idx, // edge_index row 1
    const float* __restrict__ probs,
    const v2f* __restrict__ params,        // edge_params rows
    float* __restrict__ pq_flow,           // interleaved [p,q] per node
    int E)
{
    const int stride = gridDim.x * blockDim.x;
    for (int e = blockIdx.x * blockDim.x + threadIdx.x; e < E; e += stride) {
        const int en = e + stride;
        if (en < E) {
            __builtin_prefetch(&src_idx[en], 0, 0);
            __builtin_prefetch(&dst_idx[en], 0, 0);
            __builtin_prefetch(&probs[en],   0, 0);
            __builtin_prefetch(&params[en],  0, 0);
        }
        const long long s = __builtin_nontemporal_load(&src_idx[e]);
        const long long d = __builtin_nontemporal_load(&dst_idx[e]);

        const float2 f = nf2[2 * s];                 // node_features[s, 0:2]
        const float v2 = f.x * f.x + f.y * f.y;

        const v2f   par = __builtin_nontemporal_load(&params[e]);
        const float pr  = __builtin_nontemporal_load(&probs[e]);
        const float pe  = v2 / (par[0] + EPS_F) * pr;
        const float qe  = v2 / (par[1] + EPS_F) * pr;

        atomicAdd(&pq_flow[2 * s],     pe);
        atomicAdd(&pq_flow[2 * s + 1], qe);
        if (s != d) {                                 // self-loop: dst side zeroed
            atomicAdd(&pq_flow[2 * d],     pe);
            atomicAdd(&pq_flow[2 * d + 1], qe);
        }
    }
}

// ---------------------------------------------------------------------------
// Node pass: per-node loss contributions (one b128 load per node row + one
// b64 load of its interleaved flows), then a wave32 reduction done with
// V_WMMA_F32_16X16X4_F32 (A = ones 16x4 => D[m][n] = sum_k B[k][n]), finished
// with 4 compile-time ds_swizzle xor steps, then one f32 atomic per wave.
// EXEC is all-ones at the WMMA (full blocks, bounds checked inside the loop).
// ---------------------------------------------------------------------------
__global__ __launch_bounds__(256) void pf_node_kernel(
    const float4* __restrict__ nf4,        // node_features rows (16B aligned)
    const float2* __restrict__ pq_flow,    // interleaved [p,q] per node
    float* __restrict__ acc,
    int N)
{
    float local = 0.0f;
    const int stride = gridDim.x * blockDim.x;
    for (int i = blockIdx.x * blockDim.x + threadIdx.x; i < N; i += stride) {
        const float4 nf = nf4[i];           // (v0, v1, p_load, q_load)
        const float2 pq = pq_flow[i];

        const float pb = nf.z + pq.x;
        const float qb = nf.w + pq.y;
        float c = pb * pb + qb * qb;

        const float vmag = sqrtf(nf.x * nf.x + nf.y * nf.y);
        const float lo = fmaxf(V_LO_F - vmag, 0.0f);
        const float hi = fmaxf(vmag - V_HI_F, 0.0f);
        c += lo * lo + hi * hi;

        local += c;
    }

    // Wave-level sum via WMMA: B holds the 32 lane partials (B.y rows = 0),
    // A = ones -> every D row holds the 16 column sums (acc(n) + acc(n+16)).
    v8f cmat = {};
    v2f a; a[0] = 1.0f; a[1] = 1.0f;
    v2f b; b[0] = local; b[1] = 0.0f;
    cmat = __builtin_amdgcn_wmma_f32_16x16x4_f32(
        /*neg_a=*/false, a, /*neg_b=*/false, b,
        /*c_mod=*/(short)0, cmat, /*reuse_a=*/false, /*reuse_b=*/false);

    // Sum the 16 column sums: group-of-32 ds_swizzle xor steps
    // (offset = xor_mask<<10 | and_mask 0x1F, per CDNA5 LDS swizzle encoding).
    float wsum = cmat[0];
    wsum += __int_as_float(__builtin_amdgcn_ds_swizzle(__float_as_int(wsum), 0x041F));
    wsum += __int_as_float(__builtin_amdgcn_ds_swizzle(__float_as_int(wsum), 0x081F));
    wsum += __int_as_float(__builtin_amdgcn_ds_swizzle(__float_as_int(wsum), 0x101F));
    wsum += __int_as_float(__builtin_amdgcn_ds_swizzle(__float_as_int(wsum), 0x201F));

    if ((threadIdx.x & 31) == 0) atomicAdd(acc, wsum);
}

// ---------------------------------------------------------------------------
// Finalize: both means share denominator N -> out = acc / N.
// ---------------------------------------------------------------------------
__global__ void pf_finalize_kernel(const float* __restrict__ acc,
                                   float* __restrict__ out, float invN) {
    out[0] = acc[0] * invN;
}

extern "C" void kernel_launch(void* const* d_in, const int* in_sizes, int n_in,
                              void* d_out, int out_size, void* d_ws, size_t ws_size,
                              hipStream_t stream) {
    const float*     nf     = (const float*)d_in[0];      // (N,4) f32
    const long long* eidx   = (const long long*)d_in[1];  // (2,E) i64
    const float*     probs  = (const float*)d_in[2];      // (E,)  f32
    const float*     params = (const float*)d_in[3];      // (E,2) f32

    const int N = in_sizes[0] / 4;
    const int E = in_sizes[2];

    float* ws      = (float*)d_ws;
    float* pq_flow = ws;           // 2*N floats, interleaved [p,q]
    float* acc     = ws + 2 * N;   // 1 float

    // 1) zero pq_flow and acc
    const int zn = 2 * N + 1;
    pf_zero_kernel<<<(zn + 255) / 256, 256, 0, stream>>>(ws, zn);

    // 2) edge scatter pass (grid-stride, ~4 edges/thread)
    int eblocks = (E + 1023) / 1024;
    if (eblocks < 1) eblocks = 1;
    pf_edge_kernel<<<eblocks, 256, 0, stream>>>(
        (const float2*)nf, eidx, eidx + E, probs, (const v2f*)params,
        pq_flow, E);

    // 3) node loss + reduction
    int nblocks = (N + 255) / 256;
    if (nblocks < 1) nblocks = 1;
    pf_node_kernel<<<nblocks, 256, 0, stream>>>(
        (const float4*)nf, (const float2*)pq_flow, acc, N);

    // 4) finalize scalar
    pf_finalize_kernel<<<1, 1, 0, stream>>>(acc, (float*)d_out, 1.0f / (float)N);
}